// MultiHeadAttention_90890097918184
// MI455X (gfx1250) — compile-verified
//
#include <hip/hip_runtime.h>
#include <math.h>

// ---------------------------------------------------------------------------
// Graph multi-head attention (DGL-style edge softmax) for MI455X / gfx1250.
//   q = query@Wq^T+bq ; k = key@Wk^T+bk ; v = value@Wv^T+bv     (f32 WMMA)
//   score[e,h] = dot(q[dst],k[src]) / sqrt(DK)                  (wave/edge)
//   edge softmax over incoming edges (atomic max + atomic sum)
//   agg[dst] += ex * v[src] ; out = (agg/denom)@Wo^T + bo       (f32 WMMA)
// ---------------------------------------------------------------------------

typedef __attribute__((ext_vector_type(2))) float v2f;
typedef __attribute__((ext_vector_type(8))) float v8f;

#define GDIM 64
#define GH 4
#define GDK 16

// Order-preserving uint encoding of float so segment-max can be one
// global_atomic_max_u32 (no CAS loop). encode(-inf) == 0x007FFFFF.
__device__ __forceinline__ unsigned encodeOrd(float x) {
    unsigned u = __float_as_uint(x);
    return (u & 0x80000000u) ? ~u : (u | 0x80000000u);
}
__device__ __forceinline__ float decodeOrd(unsigned k) {
    return (k & 0x80000000u) ? __uint_as_float(k ^ 0x80000000u)
                             : __uint_as_float(~k);
}

// ---------------------------------------------------------------------------
// Y[n][j] = sum_d X[n][d] * W[j][d] + bias[j]   (N x 64) @ (64 x 64)^T
// One wave computes a 16-row x 64-col stripe with V_WMMA_F32_16X16X4_F32.
// blockDim = 256 (8 waves) -> 128 rows per block.
// ---------------------------------------------------------------------------
__global__ void proj_gemm_kernel(const float* __restrict__ X,
                                 const float* __restrict__ W,
                                 const float* __restrict__ bias,
                                 float* __restrict__ Y, int N) {
    const int lane = threadIdx.x & 31;
    const int wave = threadIdx.x >> 5;
    const int half = lane >> 4;   // 0: lanes 0-15, 1: lanes 16-31
    const int lm   = lane & 15;
    const int rowBase = (blockIdx.x * 8 + wave) * 16;

    int arow = rowBase + lm;
    if (arow >= N) arow = N - 1;  // clamp loads; stores guarded (EXEC stays full)

    v8f acc0 = {}, acc1 = {}, acc2 = {}, acc3 = {};

#pragma unroll
    for (int kt = 0; kt < 16; ++kt) {
        const int k0 = kt * 4 + half * 2;  // A 16x4 f32 layout: K={0,1}|{2,3}
        v2f a  = *(const v2f*)(X + (size_t)arow * GDIM + k0);
        // B[k][n] = W[n][k]: lane's column n = lm reads contiguous W row pair.
        v2f b0 = *(const v2f*)(W + (size_t)(0 * 16 + lm) * GDIM + k0);
        v2f b1 = *(const v2f*)(W + (size_t)(1 * 16 + lm) * GDIM + k0);
        v2f b2 = *(const v2f*)(W + (size_t)(2 * 16 + lm) * GDIM + k0);
        v2f b3 = *(const v2f*)(W + (size_t)(3 * 16 + lm) * GDIM + k0);
        acc0 = __builtin_amdgcn_wmma_f32_16x16x4_f32(false, a, false, b0, (short)0, acc0, false, false);
        acc1 = __builtin_amdgcn_wmma_f32_16x16x4_f32(false, a, false, b1, (short)0, acc1, false, false);
        acc2 = __builtin_amdgcn_wmma_f32_16x16x4_f32(false, a, false, b2, (short)0, acc2, false, false);
        acc3 = __builtin_amdgcn_wmma_f32_16x16x4_f32(false, a, false, b3, (short)0, acc3, false, false);
    }

    v8f accs[4] = {acc0, acc1, acc2, acc3};
#pragma unroll
    for (int t = 0; t < 4; ++t) {
        const int col = t * 16 + lm;
        const float bj = bias[col];
#pragma unroll
        for (int r = 0; r < 8; ++r) {
            const int m = rowBase + r + half * 8;  // C/D layout: row = r + 8*half
            if (m < N) Y[(size_t)m * GDIM + col] = accs[t][r] + bj;
        }
    }
}

// ---------------------------------------------------------------------------
// OUT[n][j] = sum_d (AGG[n][d]/DEN[n][d/16]) * Wo[j][d] + bo[j]
// Same WMMA structure; softmax normalization fused into the A-fragment load.
// ---------------------------------------------------------------------------
__global__ void out_gemm_kernel(const float* __restrict__ AGG,
                                const float* __restrict__ DEN,
                                const float* __restrict__ W,
                                const float* __restrict__ bias,
                                float* __restrict__ OUT, int N) {
    const int lane = threadIdx.x & 31;
    const int wave = threadIdx.x >> 5;
    const int half = lane >> 4;
    const int lm   = lane & 15;
    const int rowBase = (blockIdx.x * 8 + wave) * 16;

    int arow = rowBase + lm;
    if (arow >= N) arow = N - 1;

    v8f acc0 = {}, acc1 = {}, acc2 = {}, acc3 = {};

#pragma unroll
    for (int kt = 0; kt < 16; ++kt) {
        const int k0 = kt * 4 + half * 2;
        // k0 is even, head boundary at multiples of 16 -> k0 and k0+1 share head
        const float dn  = DEN[(size_t)arow * GH + (k0 >> 4)];
        const float inv = (dn > 0.0f) ? (1.0f / dn) : 0.0f;
        v2f a = *(const v2f*)(AGG + (size_t)arow * GDIM + k0);
        a.x *= inv;
        a.y *= inv;
        v2f b0 = *(const v2f*)(W + (size_t)(0 * 16 + lm) * GDIM + k0);
        v2f b1 = *(const v2f*)(W + (size_t)(1 * 16 + lm) * GDIM + k0);
        v2f b2 = *(const v2f*)(W + (size_t)(2 * 16 + lm) * GDIM + k0);
        v2f b3 = *(const v2f*)(W + (size_t)(3 * 16 + lm) * GDIM + k0);
        acc0 = __builtin_amdgcn_wmma_f32_16x16x4_f32(false, a, false, b0, (short)0, acc0, false, false);
        acc1 = __builtin_amdgcn_wmma_f32_16x16x4_f32(false, a, false, b1, (short)0, acc1, false, false);
        acc2 = __builtin_amdgcn_wmma_f32_16x16x4_f32(false, a, false, b2, (short)0, acc2, false, false);
        acc3 = __builtin_amdgcn_wmma_f32_16x16x4_f32(false, a, false, b3, (short)0, acc3, false, false);
    }

    v8f accs[4] = {acc0, acc1, acc2, acc3};
#pragma unroll
    for (int t = 0; t < 4; ++t) {
        const int col = t * 16 + lm;
        const float bj = bias[col];
#pragma unroll
        for (int r = 0; r < 8; ++r) {
            const int m = rowBase + r + half * 8;
            if (m < N) OUT[(size_t)m * GDIM + col] = accs[t][r] + bj;
        }
    }
}

// ---------------------------------------------------------------------------
// Init segment-max (encoded -inf), denom, agg.
// ---------------------------------------------------------------------------
__global__ void init_kernel(unsigned* __restrict__ mEnc,
                            float* __restrict__ denom,
                            float* __restrict__ agg, int N) {
    const int i = blockIdx.x * blockDim.x + threadIdx.x;
    if (i < N * GH) {
        mEnc[i]  = 0x007FFFFFu;  // encodeOrd(-inf)
        denom[i] = 0.0f;
    }
    if (i < N * GDIM) agg[i] = 0.0f;
}

// ---------------------------------------------------------------------------
// One wave per edge: score[e,h] = dot(q[dst],k[src])/sqrt(DK); segment max.
// Lane handles dims 2l,2l+1; head = lane/8; 3x shfl_xor reduces 8-lane groups.
// ---------------------------------------------------------------------------
__global__ void edge_score_max_kernel(const float* __restrict__ qp,
                                      const float* __restrict__ kp,
                                      const int* __restrict__ src,
                                      const int* __restrict__ dst,
                                      float* __restrict__ scores,
                                      unsigned* __restrict__ mEnc, int E) {
    const int e = blockIdx.x * (blockDim.x >> 5) + (threadIdx.x >> 5);
    if (e >= E) return;
    const int lane = threadIdx.x & 31;
    const int s = src[e];
    const int d = dst[e];

    v2f q2 = *(const v2f*)(qp + (size_t)d * GDIM + lane * 2);
    v2f k2 = *(const v2f*)(kp + (size_t)s * GDIM + lane * 2);
    float p = q2.x * k2.x + q2.y * k2.y;
    p += __shfl_xor(p, 1, 32);
    p += __shfl_xor(p, 2, 32);
    p += __shfl_xor(p, 4, 32);  // lanes 0,8,16,24 hold per-head sums

    if ((lane & 7) == 0) {
        const int   h  = lane >> 3;
        const float sc = p * 0.25f;  // 1/sqrt(16)
        scores[(size_t)e * GH + h] = sc;
        atomicMax(&mEnc[(size_t)d * GH + h], encodeOrd(sc));
    }
}

// ---------------------------------------------------------------------------
// One wave per edge: ex = exp(score - m[dst]); denom += ex; agg += ex*v[src].
// Normalization by denom deferred to the output GEMM.
// ---------------------------------------------------------------------------
__global__ void edge_exp_agg_kernel(const float* __restrict__ vp,
                                    const int* __restrict__ src,
                                    const int* __restrict__ dst,
                                    const float* __restrict__ scores,
                                    const unsigned* __restrict__ mEnc,
                                    float* __restrict__ denom,
                                    float* __restrict__ agg, int E) {
    const int e = blockIdx.x * (blockDim.x >> 5) + (threadIdx.x >> 5);
    if (e >= E) return;
    const int lane = threadIdx.x & 31;
    const int s = src[e];
    const int d = dst[e];
    const int h = lane >> 3;

    const float sc = scores[(size_t)e * GH + h];
    const float m  = decodeOrd(mEnc[(size_t)d * GH + h]);
    const float ex = __expf(sc - m);

    if ((lane & 7) == 0) atomicAdd(&denom[(size_t)d * GH + h], ex);

    v2f v2 = *(const v2f*)(vp + (size_t)s * GDIM + lane * 2);
    atomicAdd(&agg[(size_t)d * GDIM + lane * 2 + 0], ex * v2.x);
    atomicAdd(&agg[(size_t)d * GDIM + lane * 2 + 1], ex * v2.y);
}

// ---------------------------------------------------------------------------
static inline size_t alignUp(size_t x, size_t a) { return (x + a - 1) & ~(a - 1); }

extern "C" void kernel_launch(void* const* d_in, const int* in_sizes, int n_in,
                              void* d_out, int out_size, void* d_ws, size_t ws_size,
                              hipStream_t stream) {
    const float* query = (const float*)d_in[0];
    const float* key_  = (const float*)d_in[1];
    const float* value = (const float*)d_in[2];
    const int*   src   = (const int*)d_in[3];
    const int*   dst   = (const int*)d_in[4];
    const float* Wq = (const float*)d_in[5];
    const float* bq = (const float*)d_in[6];
    const float* Wk = (const float*)d_in[7];
    const float* bk = (const float*)d_in[8];
    const float* Wv = (const float*)d_in[9];
    const float* bv = (const float*)d_in[10];
    const float* Wo = (const float*)d_in[11];
    const float* bo = (const float*)d_in[12];

    const int N = in_sizes[0] / GDIM;
    const int E = in_sizes[3];

    // Workspace layout (all 256B aligned for clean b64/b128 access)
    char* ws = (char*)d_ws;
    size_t off = 0;
    float* qp = (float*)(ws + off); off = alignUp(off + (size_t)N * GDIM * 4, 256);
    float* kp = (float*)(ws + off); off = alignUp(off + (size_t)N * GDIM * 4, 256);
    float* vp = (float*)(ws + off); off = alignUp(off + (size_t)N * GDIM * 4, 256);
    float* scores = (float*)(ws + off); off = alignUp(off + (size_t)E * GH * 4, 256);
    unsigned* mEnc = (unsigned*)(ws + off); off = alignUp(off + (size_t)N * GH * 4, 256);
    float* denom = (float*)(ws + off); off = alignUp(off + (size_t)N * GH * 4, 256);
    float* agg = (float*)(ws + off); off = alignUp(off + (size_t)N * GDIM * 4, 256);

    const dim3 blk(256);
    const int gemmGrid = (N + 127) / 128;   // 8 waves x 16 rows per block
    const int edgeGrid = (E + 7) / 8;       // 8 waves = 8 edges per block
    const int initGrid = (N * GDIM + 255) / 256;

    proj_gemm_kernel<<<gemmGrid, blk, 0, stream>>>(query, Wq, bq, qp, N);
    proj_gemm_kernel<<<gemmGrid, blk, 0, stream>>>(key_,  Wk, bk, kp, N);
    proj_gemm_kernel<<<gemmGrid, blk, 0, stream>>>(value, Wv, bv, vp, N);
    init_kernel<<<initGrid, blk, 0, stream>>>(mEnc, denom, agg, N);
    edge_score_max_kernel<<<edgeGrid, blk, 0, stream>>>(qp, kp, src, dst, scores, mEnc, E);
    edge_exp_agg_kernel<<<edgeGrid, blk, 0, stream>>>(vp, src, dst, scores, mEnc, denom, agg, E);
    out_gemm_kernel<<<gemmGrid, blk, 0, stream>>>(agg, denom, Wo, bo, (float*)d_out, N);
    (void)n_in; (void)out_size; (void)ws_size;
}